// SineGordonIntegrator_72018011619901
// MI455X (gfx1250) — compile-verified
//
#include <hip/hip_runtime.h>
#include <math.h>

// ---------------------------------------------------------------------------
// Sine-Gordon RK4 integrator, 2048x2048 padded grid, 15 steps.
// Bandwidth-bound stencil; whole working set (5 fields x 16.8MB = 84MB) is
// L2-resident on MI455X (192MB L2). Data movement uses both CDNA5 async
// engines: TDM (tensor_load_to_lds + s_wait_tensorcnt) for 2D halo tiles in
// k_g12/k_g4uv, and per-lane global_load_async_to_lds_b32 + s_wait_asynccnt
// in k_g3.
// ---------------------------------------------------------------------------

#define GN 2048                 // padded grid (NX+2)
#define GNN ((size_t)GN * (size_t)GN)
#define TILE 32                 // output tile per block
#define HT 34                   // max halo tile edge (TILE + 2)
#define HE (HT * HT)            // 1156 elements per halo tile
#define NTHREADS 256            // 8 wave32s

#define DT_F   0.002f
#define HDT_F  (0.5f * DT_F)

static constexpr double DX_D = 14.0 / 2047.0;
static constexpr float INV_DX2 = (float)(1.0 / (DX_D * DX_D));

typedef unsigned int u32x4 __attribute__((ext_vector_type(4)));
typedef int          i32x4 __attribute__((ext_vector_type(4)));
typedef int          i32x8 __attribute__((ext_vector_type(8)));

// ---------------------------------------------------------------------------
// TDM: one descriptor DMA-loads an HxW f32 tile (row stride GN elements) from
// global into LDS, packed with row stride W. Issued by ONE wave per block
// (TDM ignores EXEC; duplicate issues would duplicate the DMA).
// ---------------------------------------------------------------------------
__device__ __forceinline__ void tdm_tile_load(const float* src, float* lds,
                                              int rs, int cs, int H, int W) {
  unsigned long long ga = (unsigned long long)(src + (size_t)rs * GN + cs);
  unsigned lds_off = (unsigned)(unsigned long long)lds;  // low 32 bits = LDS offset

  u32x4 g0;
  g0.x = 1u;                                   // count=1 (valid user descriptor)
  g0.y = lds_off;                              // lds_addr (bytes)
  g0.z = (unsigned)(ga & 0xFFFFFFFFu);         // global_addr[31:0]
  g0.w = (unsigned)((ga >> 32) & 0x01FFFFFFu)  // global_addr[56:32]
       | (2u << 30);                           // type = 2 ("image")

  unsigned td0 = (unsigned)W, td1 = (unsigned)H;       // tensor dims == tile dims
  unsigned long long st0 = (unsigned long long)GN;     // dim0 stride (elements)
  i32x8 g1;
  g1[0] = (int)(2u << 16);                             // data_size = 2 (4 bytes)
  g1[1] = (int)((td0 & 0xFFFFu) << 16);                // tensor_dim0[15:0]
  g1[2] = (int)((td0 >> 16) | ((td1 & 0xFFFFu) << 16));// td0[31:16] | td1[15:0]
  g1[3] = (int)((td1 >> 16) | ((unsigned)W << 16));    // td1[31:16] | tile_dim0
  g1[4] = (int)((unsigned)H & 0xFFFFu);                // tile_dim1 (tile_dim2 = 0)
  g1[5] = (int)(st0 & 0xFFFFFFFFu);                    // tensor_dim0_stride[31:0]
  g1[6] = (int)((st0 >> 32) & 0xFFFFu);                // stride[47:32] | d1stride lo = 0
  g1[7] = 0;                                           // tensor_dim1_stride hi = 0

  // Benign higher-dim groups (builtin always passes them): dims = 1, strides 0.
  i32x4 g2; g2[0] = 1; g2[1] = 1; g2[2] = 0; g2[3] = 0;       // tensor_dim2/3=1
  i32x4 g3; g3[0] = 0; g3[1] = (int)(1u << 16); g3[2] = 0; g3[3] = 0; // tensor_dim4=1

#if __clang_major__ >= 23
  i32x8 gz = {};
  __builtin_amdgcn_tensor_load_to_lds(g0, g1, g2, g3, gz, 0);
#else
  __builtin_amdgcn_tensor_load_to_lds(g0, g1, g2, g3, 0);
#endif
}

__device__ __forceinline__ void wait_tensor() {
  __builtin_amdgcn_s_wait_tensorcnt(0);
}

// ---------------------------------------------------------------------------
// Per-lane CDNA5 async global->LDS loader (ASYNCcnt engine), fixed 34-stride
// layout with index-clamped halo (clamped values are never consumed).
// ---------------------------------------------------------------------------
__device__ __forceinline__ void async_tile_load(const float* __restrict__ src,
                                                float* lds, int r0, int c0) {
  for (int e = (int)threadIdx.x; e < HE; e += NTHREADS) {
    int ly = e / HT;
    int lx = e - ly * HT;
    int gr = r0 - 1 + ly; gr = gr < 0 ? 0 : (gr > GN - 1 ? GN - 1 : gr);
    int gc = c0 - 1 + lx; gc = gc < 0 ? 0 : (gc > GN - 1 ? GN - 1 : gc);
    const float* gp = src + (size_t)gr * GN + gc;
    unsigned lds_addr = (unsigned)(unsigned long long)(lds + e);
    asm volatile("global_load_async_to_lds_b32 %0, %1, off"
                 :: "v"(lds_addr), "v"((unsigned long long)gp)
                 : "memory");
  }
}

__device__ __forceinline__ void wait_async_lds() {
  asm volatile("s_wait_asynccnt 0" ::: "memory");
}

// 5-point Laplacian from an LDS tile with runtime row stride W.
__device__ __forceinline__ float lap_at_w(const float* t, int c_idx, int W) {
  float c = t[c_idx];
  float rr = (t[c_idx - W] + t[c_idx + W] - 2.0f * c) * INV_DX2;
  float cc = (t[c_idx - 1] + t[c_idx + 1] - 2.0f * c) * INV_DX2;
  return rr + cc;
}

// Clamped tile geometry for a block at (r0,c0): loaded rows rs..rs+H-1 etc.
__device__ __forceinline__ void tile_geom(int r0, int c0,
                                          int& rs, int& cs, int& H, int& W) {
  rs = (r0 == 0) ? 0 : r0 - 1;
  cs = (c0 == 0) ? 0 : c0 - 1;
  int re = r0 + TILE; if (re > GN - 1) re = GN - 1;
  int ce = c0 + TILE; if (ce > GN - 1) ce = GN - 1;
  H = re - rs + 1;
  W = ce - cs + 1;
}

// ---------------------------------------------------------------------------
// Pass 1+2 fused (TDM loads): g1 = G(u), g2 = G(u + 0.5*DT*v)
// G(w) = lap_interior(w) - sin(w)   (c2 = m = 1)
// ---------------------------------------------------------------------------
__global__ void __launch_bounds__(NTHREADS)
k_g12(const float* __restrict__ u, const float* __restrict__ v,
      float* g1, float* g2) {
  __shared__ float su[HE];
  __shared__ float sv[HE];
  const int r0 = blockIdx.y * TILE, c0 = blockIdx.x * TILE;
  int rs, cs, H, W;
  tile_geom(r0, c0, rs, cs, H, W);
  if (threadIdx.x < 32) {                 // single wave drives the TDM
    tdm_tile_load(u, su, rs, cs, H, W);
    tdm_tile_load(v, sv, rs, cs, H, W);
    wait_tensor();
  }
  __syncthreads();
  const int cnt = H * W;
  for (int e = (int)threadIdx.x; e < cnt; e += NTHREADS)   // sv <- w2
    sv[e] = su[e] + HDT_F * sv[e];
  __syncthreads();

  const int tx = threadIdx.x & 31, ty = threadIdx.x >> 5;
  const int ro = r0 - rs, co = c0 - cs;   // 0 or 1
#pragma unroll
  for (int k = 0; k < 4; ++k) {
    const int lyo = ty + 8 * k;
    const int row = r0 + lyo, col = c0 + tx;
    const int ci = (lyo + ro) * W + tx + co;
    const bool interior = (row >= 1) && (row <= GN - 2) &&
                          (col >= 1) && (col <= GN - 2);
    float uc = su[ci];
    float wc = sv[ci];
    float lu = 0.0f, lw = 0.0f;
    if (interior) { lu = lap_at_w(su, ci, W); lw = lap_at_w(sv, ci, W); }
    size_t idx = (size_t)row * GN + col;
    g1[idx] = lu - sinf(uc);
    g2[idx] = lw - sinf(wc);
  }
}

// ---------------------------------------------------------------------------
// Pass 3 (per-lane async loads): g3 = G(u + 0.5*DT*(v + 0.5*DT*g1))
// ---------------------------------------------------------------------------
__global__ void __launch_bounds__(NTHREADS)
k_g3(const float* __restrict__ u, const float* __restrict__ v,
     const float* __restrict__ g1, float* g3) {
  __shared__ float su[HE];
  __shared__ float sv[HE];
  __shared__ float sg[HE];
  const int r0 = blockIdx.y * TILE, c0 = blockIdx.x * TILE;
  async_tile_load(u, su, r0, c0);
  async_tile_load(v, sv, r0, c0);
  async_tile_load(g1, sg, r0, c0);
  wait_async_lds();
  __syncthreads();
  for (int e = (int)threadIdx.x; e < HE; e += NTHREADS)    // su <- w3
    su[e] = su[e] + HDT_F * (sv[e] + HDT_F * sg[e]);
  __syncthreads();

  const int tx = threadIdx.x & 31, ty = threadIdx.x >> 5;
#pragma unroll
  for (int k = 0; k < 4; ++k) {
    const int lyo = ty + 8 * k;
    const int row = r0 + lyo, col = c0 + tx;
    const int ci = (lyo + 1) * HT + tx + 1;
    const bool interior = (row >= 1) && (row <= GN - 2) &&
                          (col >= 1) && (col <= GN - 2);
    float wc = su[ci];
    float lw = interior ? lap_at_w(su, ci, HT) : 0.0f;
    g3[(size_t)row * GN + col] = lw - sinf(wc);
  }
}

// ---------------------------------------------------------------------------
// Pass 4 fused with the state update (TDM loads):
//   g4 = G(u + DT*(v + DT*g3)); un/vn per classic RK4 combination.
// g1/g2 are read pointwise -> un/vn may alias the g1/g2 buffers.
// ---------------------------------------------------------------------------
__global__ void __launch_bounds__(NTHREADS)
k_g4uv(const float* __restrict__ u, const float* __restrict__ v,
       const float* g1, const float* g2, const float* __restrict__ g3,
       float* un, float* vn) {
  __shared__ float su[HE];
  __shared__ float sv[HE];
  __shared__ float sg[HE];
  const int r0 = blockIdx.y * TILE, c0 = blockIdx.x * TILE;
  int rs, cs, H, W;
  tile_geom(r0, c0, rs, cs, H, W);
  if (threadIdx.x < 32) {
    tdm_tile_load(u,  su, rs, cs, H, W);
    tdm_tile_load(v,  sv, rs, cs, H, W);
    tdm_tile_load(g3, sg, rs, cs, H, W);
    wait_tensor();
  }
  __syncthreads();

  const int tx = threadIdx.x & 31, ty = threadIdx.x >> 5;
  const int ro = r0 - rs, co = c0 - cs;
  float g3c[4];
#pragma unroll
  for (int k = 0; k < 4; ++k)             // capture centers before sg overwrite
    g3c[k] = sg[(ty + 8 * k + ro) * W + tx + co];
  __syncthreads();
  const int cnt = H * W;
  for (int e = (int)threadIdx.x; e < cnt; e += NTHREADS)   // sg <- w4
    sg[e] = su[e] + DT_F * (sv[e] + DT_F * sg[e]);
  __syncthreads();

#pragma unroll
  for (int k = 0; k < 4; ++k) {
    const int lyo = ty + 8 * k;
    const int row = r0 + lyo, col = c0 + tx;
    const int ci = (lyo + ro) * W + tx + co;
    const bool interior = (row >= 1) && (row <= GN - 2) &&
                          (col >= 1) && (col <= GN - 2);
    float uc = su[ci];
    float vc = sv[ci];
    float wc = sg[ci];
    float lw = interior ? lap_at_w(sg, ci, W) : 0.0f;
    float g4 = lw - sinf(wc);

    size_t idx = (size_t)row * GN + col;
    float k1v = DT_F * g1[idx];
    float k2v = DT_F * g2[idx];
    float k3v = DT_F * g3c[k];
    float k4v = DT_F * g4;
    float k1u = DT_F * vc;
    float k2u = DT_F * (vc + 0.5f * k1v);
    float k3u = DT_F * (vc + 0.5f * k2v);
    float k4u = DT_F * (vc + k3v);
    un[idx] = uc + (k1u + 2.0f * k2u + 2.0f * k3u + k4u) / 6.0f;
    vn[idx] = vc + (k1v + 2.0f * k2v + 2.0f * k3v + k4v) / 6.0f;
  }
}

// ---------------------------------------------------------------------------
// Boundary conditions (race-free; reproduces the reference's sequential edge
// assignment order, corners substituted analytically).
// ---------------------------------------------------------------------------
__global__ void k_bc(float* u, float* v) {
  int t = blockIdx.x * blockDim.x + threadIdx.x;
  if (t >= GN) return;
  if (t >= 1 && t <= GN - 2) {
    u[t] = u[GN + t];                                        // row 0 <- row 1
    u[(size_t)(GN - 1) * GN + t] = u[(size_t)(GN - 2) * GN + t];
  }
  float lft, rgt;
  if (t == 0)            { lft = u[GN + 1];                    rgt = u[GN + (GN - 2)]; }
  else if (t == GN - 1)  { lft = u[(size_t)(GN - 2) * GN + 1]; rgt = u[(size_t)(GN - 2) * GN + (GN - 2)]; }
  else                   { lft = u[(size_t)t * GN + 1];        rgt = u[(size_t)t * GN + (GN - 2)]; }
  u[(size_t)t * GN]            = lft;
  u[(size_t)t * GN + (GN - 1)] = rgt;
  if (t >= 1 && t <= GN - 2) {
    v[t] = 0.0f;
    v[(size_t)(GN - 1) * GN + t] = 0.0f;
    v[(size_t)t * GN] = 0.0f;
    v[(size_t)t * GN + (GN - 1)] = 0.0f;
  }
}

// ---------------------------------------------------------------------------
// Host launcher: 15 RK4 steps, 5-buffer rotation; d_out slots double as
// scratch when ws is small (then a final async d2d copy places the result).
// ---------------------------------------------------------------------------
extern "C" void kernel_launch(void* const* d_in, const int* in_sizes, int n_in,
                              void* d_out, int out_size, void* d_ws, size_t ws_size,
                              hipStream_t stream) {
  (void)in_sizes; (void)n_in; (void)out_size;
  const float* u0 = (const float*)d_in[0];
  const float* v0 = (const float*)d_in[1];
  float* out = (float*)d_out;
  float* wsf = (float*)d_ws;

  const bool ws5 = ws_size >= 5 * GNN * sizeof(float);
  float* W[5];
  W[0] = wsf; W[1] = wsf + GNN; W[2] = wsf + 2 * GNN;
  if (ws5) { W[3] = wsf + 3 * GNN; W[4] = wsf + 4 * GNN; }
  else     { W[3] = out;           W[4] = out + GNN;     }

  dim3 grid(GN / TILE, GN / TILE), block(NTHREADS);
  dim3 egrid(GN / NTHREADS), eblock(NTHREADS);

  float* G3 = W[2];
  float* lastU = nullptr; float* lastV = nullptr;
  for (int s = 0; s < 15; ++s) {
    const float *U, *V; float *G1, *G2;
    if (s == 0)      { U = u0;   V = v0;   G1 = W[0]; G2 = W[1]; }
    else if (s & 1)  { U = W[0]; V = W[1]; G1 = W[3]; G2 = W[4]; }
    else             { U = W[3]; V = W[4]; G1 = W[0]; G2 = W[1]; }
    if (ws5 && s == 14) { G1 = out; G2 = out + GNN; }  // final step lands in d_out

    k_g12 <<<grid, block, 0, stream>>>(U, V, G1, G2);
    k_g3  <<<grid, block, 0, stream>>>(U, V, G1, G3);
    k_g4uv<<<grid, block, 0, stream>>>(U, V, G1, G2, G3, G1, G2); // un->G1, vn->G2
    k_bc  <<<egrid, eblock, 0, stream>>>(G1, G2);
    lastU = G1; lastV = G2;
  }

  if (!ws5) {  // final state is in W[0]/W[1] (s=14 is even); move into d_out
    hipMemcpyAsync(out,       lastU, GNN * sizeof(float), hipMemcpyDeviceToDevice, stream);
    hipMemcpyAsync(out + GNN, lastV, GNN * sizeof(float), hipMemcpyDeviceToDevice, stream);
  }
}